// RecurrentNetTimeFixed_4733053960304
// MI455X (gfx1250) — compile-verified
//
#include <hip/hip_runtime.h>

#define N_IN   100
#define N_OUT  100
#define N_HID  500
#define BATCH  256
#define LENGTH 300
#define TB     (BATCH * LENGTH)   // 76800 rows

typedef __attribute__((ext_vector_type(2))) float v2f;
typedef __attribute__((ext_vector_type(8))) float v8f;

// ---------------------------------------------------------------------------
// Kernel 1: W_comb = W_ih (500x500) @ W_in (500x100)   -> [500,100]
// One 16x16 tile per wave; K=500 via v_wmma_f32_16x16x4_f32.
// ---------------------------------------------------------------------------
__global__ void wcomb_kernel(const float* __restrict__ W_ih,
                             const float* __restrict__ W_in,
                             float* __restrict__ Wcomb) {
    const int wave = (blockIdx.x * blockDim.x + threadIdx.x) >> 5;
    const int Nt = 7;                       // ceil(100/16)
    const int mt = wave / Nt, nt = wave % Nt;
    const int m0 = mt * 16, n0 = nt * 16;
    const int lane = threadIdx.x & 31;
    const int hi = lane >> 4, lo = lane & 15;

    int ar = m0 + lo; if (ar > N_HID - 1) ar = N_HID - 1;     // clamp (discarded later)
    int nc = n0 + lo; if (nc > N_IN - 1) nc = N_IN - 1;

    const float* __restrict__ arow = W_ih + (size_t)ar * N_HID + (hi << 1);
    v8f acc = {};
    for (int k = 0; k < N_HID; k += 4) {
        v2f a = *(const v2f*)(arow + k);
        v2f b;   // B[k][n] = W_in[k][n] (column gather, small matrix)
        b.x = W_in[(size_t)(k + (hi << 1) + 0) * N_IN + nc];
        b.y = W_in[(size_t)(k + (hi << 1) + 1) * N_IN + nc];
        acc = __builtin_amdgcn_wmma_f32_16x16x4_f32(false, a, false, b,
                                                    (short)0, acc, false, false);
    }
    const int col = n0 + lo;
    if (col < N_IN) {
        for (int r = 0; r < 8; ++r) {
            const int row = m0 + r + (hi << 3);
            if (row < N_HID) Wcomb[(size_t)row * N_IN + col] = acc[r];
        }
    }
}

// ---------------------------------------------------------------------------
// Kernel 2: b_comb[i] = sum_h W_ih[i][h]*b_in[h] + b_ih[i] + b_hh[i]
// ---------------------------------------------------------------------------
__global__ void bcomb_kernel(const float* __restrict__ W_ih,
                             const float* __restrict__ b_in,
                             const float* __restrict__ b_ih,
                             const float* __restrict__ b_hh,
                             float* __restrict__ bcomb) {
    const int i = blockIdx.x * blockDim.x + threadIdx.x;
    if (i < N_HID) {
        float s = 0.0f;
        for (int h = 0; h < N_HID; ++h)
            s = fmaf(W_ih[(size_t)i * N_HID + h], b_in[h], s);
        bcomb[i] = s + b_ih[i] + b_hh[i];
    }
}

// ---------------------------------------------------------------------------
// Kernel 3: G = input (76800x100) @ W_comb^T + b_comb  -> hidden_list region
// of d_out (row (b*T+t), ld=500).  Block = one 16-row tile (blockIdx = mt);
// A rows staged in LDS; each wave computes a 16x64 strip (4 col tiles,
// 4 independent WMMA accumulator chains).
// ---------------------------------------------------------------------------
__global__ void gproj_kernel(const float* __restrict__ input,
                             const float* __restrict__ Wcomb,
                             const float* __restrict__ bcomb,
                             float* __restrict__ G) {
    __shared__ float aLds[16 * N_IN];                 // 6.4 KB
    const int m0 = blockIdx.x * 16;
    // stage 16 contiguous A rows (16*100 floats)
    {
        const float* __restrict__ src = input + (size_t)m0 * N_IN;
        for (int i = threadIdx.x; i < 16 * N_IN; i += blockDim.x)
            aLds[i] = src[i];
    }
    __syncthreads();

    const int ng = threadIdx.x >> 5;                  // wave -> 64-col group (0..7)
    const int n0 = ng * 64;
    const int lane = threadIdx.x & 31;
    const int hi = lane >> 4, lo = lane & 15;

    int ncol[4], nc[4];
    const float* __restrict__ wrow[4];
    for (int j = 0; j < 4; ++j) {
        ncol[j] = n0 + j * 16 + lo;
        nc[j] = (ncol[j] > N_HID - 1) ? (N_HID - 1) : ncol[j];
        wrow[j] = Wcomb + (size_t)nc[j] * N_IN + (hi << 1);
    }

    v8f acc0 = {}, acc1 = {}, acc2 = {}, acc3 = {};
    const int abase = lo * N_IN + (hi << 1);
    for (int k = 0; k < N_IN; k += 4) {
        v2f a  = *(const v2f*)(&aLds[abase + k]);     // ds_load_b64
        v2f b0 = *(const v2f*)(wrow[0] + k);
        v2f b1 = *(const v2f*)(wrow[1] + k);
        v2f b2 = *(const v2f*)(wrow[2] + k);
        v2f b3 = *(const v2f*)(wrow[3] + k);
        acc0 = __builtin_amdgcn_wmma_f32_16x16x4_f32(false, a, false, b0, (short)0, acc0, false, false);
        acc1 = __builtin_amdgcn_wmma_f32_16x16x4_f32(false, a, false, b1, (short)0, acc1, false, false);
        acc2 = __builtin_amdgcn_wmma_f32_16x16x4_f32(false, a, false, b2, (short)0, acc2, false, false);
        acc3 = __builtin_amdgcn_wmma_f32_16x16x4_f32(false, a, false, b3, (short)0, acc3, false, false);
    }

    v8f* accs[4] = {&acc0, &acc1, &acc2, &acc3};
    for (int j = 0; j < 4; ++j) {
        if (ncol[j] < N_HID) {
            const float bias = bcomb[ncol[j]];
            for (int r = 0; r < 8; ++r) {
                const int row = m0 + r + (hi << 3);
                G[(size_t)row * N_HID + ncol[j]] = (*accs[j])[r] + bias;
            }
        }
    }
}

// ---------------------------------------------------------------------------
// Kernel 4: one recurrence step (critical sequential path).
//   pre  = G[:,t,:] + Hcur @ W_hh^T     (G tile read from hiddenList region)
//   hnew = (1-alpha)*Hcur + alpha*relu(pre)
// Block = one 16-row batch tile (blockIdx = mt, 16 blocks); Hcur rows staged
// in LDS (32 KB); each wave: 16x64 strip = 4 independent WMMA chains to hide
// XDL latency (only ~1 wave/SIMD at this size -> ILP must come from within).
// ---------------------------------------------------------------------------
__global__ void step_kernel(const float* __restrict__ Hcur,
                            const float* __restrict__ W_hh,
                            const float* __restrict__ alpha,
                            float* __restrict__ hiddenList,
                            float* __restrict__ Hnext,
                            float* __restrict__ hiddenFinal,
                            int t) {
    __shared__ float aLds[16 * N_HID];                // 32 KB
    const int m0 = blockIdx.x * 16;
    {
        const float* __restrict__ src = Hcur + (size_t)m0 * N_HID;
        for (int i = threadIdx.x; i < 16 * N_HID; i += blockDim.x)
            aLds[i] = src[i];
    }
    __syncthreads();

    const int ng = threadIdx.x >> 5;                  // 0..7 -> 64-col group
    const int n0 = ng * 64;
    const int lane = threadIdx.x & 31;
    const int hi = lane >> 4, lo = lane & 15;

    int ncol[4], nc[4];
    const float* __restrict__ wrow[4];
    for (int j = 0; j < 4; ++j) {
        ncol[j] = n0 + j * 16 + lo;
        nc[j] = (ncol[j] > N_HID - 1) ? (N_HID - 1) : ncol[j];
        wrow[j] = W_hh + (size_t)nc[j] * N_HID + (hi << 1);
    }

    // init accumulators from the G tiles (clamped reads, extra lanes discarded)
    v8f acc0, acc1, acc2, acc3;
    v8f* accs[4] = {&acc0, &acc1, &acc2, &acc3};
    for (int j = 0; j < 4; ++j)
        for (int r = 0; r < 8; ++r) {
            const int row = m0 + r + (hi << 3);       // batch index
            (*accs[j])[r] = hiddenList[((size_t)row * LENGTH + t) * N_HID + nc[j]];
        }

    const int abase = lo * N_HID + (hi << 1);
    for (int k = 0; k < N_HID; k += 4) {
        v2f a  = *(const v2f*)(&aLds[abase + k]);     // ds_load_b64
        v2f b0 = *(const v2f*)(wrow[0] + k);
        v2f b1 = *(const v2f*)(wrow[1] + k);
        v2f b2 = *(const v2f*)(wrow[2] + k);
        v2f b3 = *(const v2f*)(wrow[3] + k);
        acc0 = __builtin_amdgcn_wmma_f32_16x16x4_f32(false, a, false, b0, (short)0, acc0, false, false);
        acc1 = __builtin_amdgcn_wmma_f32_16x16x4_f32(false, a, false, b1, (short)0, acc1, false, false);
        acc2 = __builtin_amdgcn_wmma_f32_16x16x4_f32(false, a, false, b2, (short)0, acc2, false, false);
        acc3 = __builtin_amdgcn_wmma_f32_16x16x4_f32(false, a, false, b3, (short)0, acc3, false, false);
    }

    const bool last = (t == LENGTH - 1);
    for (int j = 0; j < 4; ++j) {
        if (ncol[j] < N_HID) {
            const float al = alpha[ncol[j]];
            for (int r = 0; r < 8; ++r) {
                const int row = m0 + r + (hi << 3);
                const float hcell = fmaxf((*accs[j])[r], 0.0f);
                const float hp = aLds[(r + (hi << 3)) * N_HID + ncol[j]];  // h(t-1) from LDS
                const float hn = (1.0f - al) * hp + al * hcell;
                hiddenList[((size_t)row * LENGTH + t) * N_HID + ncol[j]] = hn;
                Hnext[(size_t)row * N_HID + ncol[j]] = hn;
                if (last) hiddenFinal[(size_t)row * N_HID + ncol[j]] = hn;
            }
        }
    }
}

// ---------------------------------------------------------------------------
// Kernel 5: output_list = hidden_list (76800x500) @ W_out^T + b_out -> [76800,100]
// ---------------------------------------------------------------------------
__global__ void outproj_kernel(const float* __restrict__ hiddenList,
                               const float* __restrict__ W_out,
                               const float* __restrict__ b_out,
                               float* __restrict__ outputList) {
    const int wave = (blockIdx.x * blockDim.x + threadIdx.x) >> 5;
    const int Nt = 7;                        // ceil(100/16)
    const int mt = wave / Nt, nt = wave % Nt;
    const int m0 = mt * 16, n0 = nt * 16;
    const int lane = threadIdx.x & 31;
    const int hi = lane >> 4, lo = lane & 15;

    const int ncol = n0 + lo;
    const int nc = (ncol > N_OUT - 1) ? (N_OUT - 1) : ncol;

    const float* __restrict__ arow = hiddenList + (size_t)(m0 + lo) * N_HID + (hi << 1);
    const float* __restrict__ wrow = W_out + (size_t)nc * N_HID + (hi << 1);
    v8f acc = {};
    for (int k = 0; k < N_HID; k += 4) {
        v2f a = *(const v2f*)(arow + k);
        v2f b = *(const v2f*)(wrow + k);
        acc = __builtin_amdgcn_wmma_f32_16x16x4_f32(false, a, false, b,
                                                    (short)0, acc, false, false);
    }
    if (ncol < N_OUT) {
        const float bias = b_out[ncol];
        for (int r = 0; r < 8; ++r) {
            const int row = m0 + r + (hi << 3);
            outputList[(size_t)row * N_OUT + ncol] = acc[r] + bias;
        }
    }
}

// ---------------------------------------------------------------------------
extern "C" void kernel_launch(void* const* d_in, const int* in_sizes, int n_in,
                              void* d_out, int out_size, void* d_ws, size_t ws_size,
                              hipStream_t stream) {
    const float* input = (const float*)d_in[0];   // [256,300,100]
    const float* h0    = (const float*)d_in[1];   // [256,500]
    const float* W_in  = (const float*)d_in[2];   // [500,100]
    const float* b_in  = (const float*)d_in[3];   // [500]
    const float* W_ih  = (const float*)d_in[4];   // [500,500]
    const float* b_ih  = (const float*)d_in[5];   // [500]
    const float* W_hh  = (const float*)d_in[6];   // [500,500]
    const float* b_hh  = (const float*)d_in[7];   // [500]
    const float* W_out = (const float*)d_in[8];   // [100,500]
    const float* b_out = (const float*)d_in[9];   // [100]
    const float* alpha = (const float*)d_in[10];  // [500,1] (@ones(1) == itself)

    float* out         = (float*)d_out;
    float* hiddenList  = out;                                   // [256*300, 500]
    float* outputList  = out + (size_t)TB * N_HID;              // [256*300, 100]
    float* hiddenFinal = outputList + (size_t)TB * N_OUT;       // [256, 500]

    float* ws    = (float*)d_ws;
    float* Wcomb = ws;                          // 500*100
    float* bcomb = Wcomb + N_HID * N_IN;        // 500 (padded to 512)
    float* Hbuf0 = bcomb + 512;                 // 256*500
    float* Hbuf1 = Hbuf0 + BATCH * N_HID;       // 256*500
    float* bufs[2] = {Hbuf0, Hbuf1};

    // h(0) = given initial hidden state
    hipMemcpyAsync(Hbuf0, h0, (size_t)BATCH * N_HID * sizeof(float),
                   hipMemcpyDeviceToDevice, stream);

    // Fold frozen input layer: W_comb = W_ih @ W_in ; b_comb = W_ih@b_in + b_ih + b_hh
    wcomb_kernel<<<28, 256, 0, stream>>>(W_ih, W_in, Wcomb);          // 224 tiles / 8 waves
    bcomb_kernel<<<2, 256, 0, stream>>>(W_ih, b_in, b_ih, b_hh, bcomb);

    // G = input @ W_comb^T + b_comb  (into hiddenList region of d_out)
    // block = one 16-row tile, 8 waves x (16x64 strip)
    gproj_kernel<<<4800, 256, 0, stream>>>(input, Wcomb, bcomb, hiddenList);

    // Sequential recurrence: 300 WMMA GEMM steps, ping-pong h buffers
    for (int t = 0; t < LENGTH; ++t) {
        step_kernel<<<16, 256, 0, stream>>>(bufs[t & 1], W_hh, alpha,
                                            hiddenList, bufs[(t + 1) & 1],
                                            hiddenFinal, t);
    }

    // output_list = hidden_list @ W_out^T + b_out
    outproj_kernel<<<4200, 256, 0, stream>>>(hiddenList, W_out, b_out, outputList); // 33600 tiles
}